// LIF_41377714930178
// MI455X (gfx1250) — compile-verified
//
#include <hip/hip_runtime.h>

// ----------------------------------------------------------------------------
// Fused LIF forward for MI455X (gfx1250, wave32).
//
// tx: [T=32, B=128, N=16384] f32 -> ty: [T, B, N] f32 spike train.
//   v_t = decay * v_{t-1} * (1 - s_{t-1}) + x_t   (rest = 0)
//   s_t = v_t > threshold
//
// 512 MB streamed once, ~0.27 GFLOP -> HBM-bound (~22 us @ 23.3 TB/s).
// Primary path: CDNA5 async DMA (GLOBAL_LOAD_ASYNC_TO_LDS_B128, ASYNCcnt)
// staging a 4-deep LDS ring per block; per-wave private slices -> no barriers,
// just s_wait_asynccnt / s_wait_dscnt. Output: direct NT b128 stores.
// Fallback path (builtin absent): register-pipelined NT b128 loads.
// ----------------------------------------------------------------------------

typedef float v4f __attribute__((ext_vector_type(4)));

namespace {
constexpr int   T       = 32;
constexpr int   BATCH   = 128;
constexpr int   NEURONS = 16384;
constexpr int   BN      = BATCH * NEURONS;  // 2,097,152 scalars per timestep
constexpr int   NV      = BN / 4;           // 524,288 float4 per timestep
constexpr float DECAY   = 0.2f;
constexpr float THRESH  = 0.3f;
constexpr int   BLOCK   = 256;              // 8 wave32 per block
constexpr int   DEPTH   = 4;                // LDS ring depth: 4 x 4KB = 16KB/block
constexpr int   PF      = 4;                // register prefetch depth (fallback)
} // namespace

// One LIF timestep on a float4. s in {0,1}; rest = 0 simplifies hard reset to
// v_t = s_{t-1} ? x_t : decay*v_{t-1} + x_t. ~3 VALU/component: fma, cndmask,
// cmp+cndmask.
__device__ __forceinline__ void lif_step(v4f& v, v4f& s, const v4f x) {
#pragma unroll
  for (int c = 0; c < 4; ++c) {
    float vdec = __builtin_fmaf(DECAY, v[c], x[c]); // no-reset candidate
    float vv   = (s[c] > 0.5f) ? x[c] : vdec;        // hard reset to rest=0
    v[c]       = vv;
    s[c]       = (vv > THRESH) ? 1.0f : 0.0f;
  }
}

#if __has_builtin(__builtin_amdgcn_global_load_async_to_lds_b128)
// ======================= CDNA5 async-to-LDS path ============================

// The builtin's prototype (from clang diagnostics) takes pointers to GNU-style
// int4 vectors: param0 = AS(1) "__device__" int4*, param1 = AS(3) int4*.
typedef int v4i_raw __attribute__((vector_size(4 * sizeof(int))));
typedef __attribute__((address_space(1))) v4i_raw gbuf_t; // global (AS1)
typedef __attribute__((address_space(3))) v4i_raw lbuf_t; // LDS (AS3)

__global__ __launch_bounds__(BLOCK) void lif_fwd_41377714930178(
    const v4f* __restrict__ in, v4f* __restrict__ out) {
  // 4-deep ring of 4 KB block tiles. Each lane only ever touches its own
  // 16 B cell per slot -> per-wave private slices, no workgroup barriers.
  __shared__ v4f tile[DEPTH][BLOCK];

  const int tid = threadIdx.x;
  const int idx = blockIdx.x * BLOCK + tid; // grid sized exactly, no bounds chk
  const v4f* __restrict__ ip = in + idx;
  v4f* __restrict__ op       = out + idx;

  // Prime DEPTH timesteps: HBM -> LDS via async DMA. No VGPR staging; each
  // issue is tracked by ASYNCcnt and completes ("done") in order per wave.
#pragma unroll
  for (int i = 0; i < DEPTH; ++i)
    __builtin_amdgcn_global_load_async_to_lds_b128(
        (gbuf_t*)(ip + i * NV), (lbuf_t*)&tile[i][tid], 0, 0);

  v4f v = {0.f, 0.f, 0.f, 0.f};
  v4f s = {0.f, 0.f, 0.f, 0.f};

  // Steady state. Before step t: issued = DEPTH + t async loads; requiring
  // outstanding <= DEPTH-1 guarantees loads 0..t have landed in LDS.
#pragma unroll
  for (int t = 0; t < T - DEPTH; ++t) {
    asm volatile("s_wait_asynccnt %0" ::"i"(DEPTH - 1) : "memory");
    v4f x = tile[t % DEPTH][tid]; // ds_load_b128
    // Close the LDS WAR hazard: the ds read must retire before the async
    // engine may overwrite this slot with timestep t+DEPTH.
    asm volatile("s_wait_dscnt 0x0" ::: "memory");
    __builtin_amdgcn_global_load_async_to_lds_b128(
        (gbuf_t*)(ip + (t + DEPTH) * NV), (lbuf_t*)&tile[t % DEPTH][tid], 0, 0);
    lif_step(v, s, x);
    __builtin_nontemporal_store(s, op + t * NV); // NT b128, write-once stream
  }

  // Tail drain: all 32 loads issued; load t done iff outstanding <= 31-t.
#define LIF_TAIL(t, w)                                          \
  {                                                             \
    asm volatile("s_wait_asynccnt %0" ::"i"(w) : "memory");     \
    v4f x = tile[(t) % DEPTH][tid];                             \
    lif_step(v, s, x);                                          \
    __builtin_nontemporal_store(s, op + (t) * NV);              \
  }
  LIF_TAIL(T - 4, 3)
  LIF_TAIL(T - 3, 2)
  LIF_TAIL(T - 2, 1)
  LIF_TAIL(T - 1, 0)
#undef LIF_TAIL
}

#else
// ================== fallback: register-pipelined direct loads ===============

__global__ __launch_bounds__(BLOCK) void lif_fwd_41377714930178(
    const v4f* __restrict__ in, v4f* __restrict__ out) {
  const int idx = blockIdx.x * BLOCK + threadIdx.x;
  const v4f* __restrict__ ip = in + idx;
  v4f* __restrict__ op       = out + idx;

  v4f buf[PF];
#pragma unroll
  for (int i = 0; i < PF; ++i)
    buf[i] = __builtin_nontemporal_load(ip + i * NV);

  v4f v = {0.f, 0.f, 0.f, 0.f};
  v4f s = {0.f, 0.f, 0.f, 0.f};

#pragma unroll
  for (int t = 0; t < T; ++t) {
    v4f x = buf[t % PF];
    if (t + PF < T)
      buf[t % PF] = __builtin_nontemporal_load(ip + (t + PF) * NV);
    lif_step(v, s, x);
    __builtin_nontemporal_store(s, op + t * NV);
  }
}

#endif

extern "C" void kernel_launch(void* const* d_in, const int* in_sizes, int n_in,
                              void* d_out, int out_size, void* d_ws,
                              size_t ws_size, hipStream_t stream) {
  (void)in_sizes; (void)n_in; (void)out_size; (void)d_ws; (void)ws_size;
  const v4f* in = reinterpret_cast<const v4f*>(d_in[0]);
  v4f* out      = reinterpret_cast<v4f*>(d_out);
  const int grid = NV / BLOCK; // 2048 blocks, exact cover of 524,288 lanes
  lif_fwd_41377714930178<<<grid, BLOCK, 0, stream>>>(in, out);
}